// SurfaceDistanceLoss_31705448579089
// MI455X (gfx1250) — compile-verified
//
#include <hip/hip_runtime.h>
#include <hip/hip_bf16.h>

// Problem constants (B, 1, D, H, W) = (2, 1, 64, 128, 128)
#define Bc   2
#define Dc   64
#define Hc   128
#define Wc   128
#define HWc  (Hc * Wc)        // 16384
#define DHWc (Dc * Hc * Wc)   // 1048576
#define NTOT (Bc * DHWc)      // 2097152
#define LARGE_F 1.0e8f

// ---------------- CDNA5 async global->LDS staging (guarded) ----------------
#if defined(__gfx1250__) && __has_builtin(__builtin_amdgcn_global_load_async_to_lds_b32)
#define HAS_ASYNC 1
#else
#define HAS_ASYNC 0
#endif

#if HAS_ASYNC
typedef __attribute__((address_space(1))) int as1_int;
typedef __attribute__((address_space(3))) int as3_int;
#endif

__device__ __forceinline__ void lds_stage_f32(const float* g, float* l) {
#if HAS_ASYNC
  __builtin_amdgcn_global_load_async_to_lds_b32(
      (as1_int*)g, (as3_int*)l, 0, 0);
#else
  *l = *g;
#endif
}

__device__ __forceinline__ void lds_stage_wait() {
#if HAS_ASYNC
#if __has_builtin(__builtin_amdgcn_s_wait_asynccnt)
  __builtin_amdgcn_s_wait_asynccnt(0);
#else
  asm volatile("s_wait_asynccnt 0" ::: "memory");
#endif
#endif
}

// ---------------- Kernel 1: masks, erosion, surface -> initial d2 ----------
// surface = m && !(all 6 face-neighbors set); OOB neighbor counts as 0.
__global__ __launch_bounds__(256) void k_init(const float* __restrict__ pred,
                                              const float* __restrict__ target,
                                              float* __restrict__ d2p,
                                              float* __restrict__ d2t) {
  int idx = blockIdx.x * 256 + threadIdx.x;
  if (idx >= NTOT) return;
  int w = idx & (Wc - 1);
  int h = (idx >> 7) & (Hc - 1);
  int d = (idx >> 14) & (Dc - 1);

  // pred mask: sigmoid(x) > 0.5  <=>  x > 0
  float op;
  if (!(pred[idx] > 0.0f)) {
    op = LARGE_F;
  } else {
    bool all = (w > 0)      && (pred[idx - 1]   > 0.0f) &&
               (w < Wc - 1) && (pred[idx + 1]   > 0.0f) &&
               (h > 0)      && (pred[idx - Wc]  > 0.0f) &&
               (h < Hc - 1) && (pred[idx + Wc]  > 0.0f) &&
               (d > 0)      && (pred[idx - HWc] > 0.0f) &&
               (d < Dc - 1) && (pred[idx + HWc] > 0.0f);
    op = all ? LARGE_F : 0.0f;   // interior (eroded) -> not surface
  }
  d2p[idx] = op;

  // target mask: t != 0
  float ot;
  if (!(target[idx] != 0.0f)) {
    ot = LARGE_F;
  } else {
    bool all = (w > 0)      && (target[idx - 1]   != 0.0f) &&
               (w < Wc - 1) && (target[idx + 1]   != 0.0f) &&
               (h > 0)      && (target[idx - Wc]  != 0.0f) &&
               (h < Hc - 1) && (target[idx + Wc]  != 0.0f) &&
               (d > 0)      && (target[idx - HWc] != 0.0f) &&
               (d < Dc - 1) && (target[idx + HWc] != 0.0f);
    ot = all ? LARGE_F : 0.0f;
  }
  d2t[idx] = ot;
}

// ---------------- Kernel 2: EDT pass along D (n=64, stride H*W) ------------
// Tile: 64 (d) x 32 (contiguous w). Blocks per array = B*H*(W/32) = 1024.
__global__ __launch_bounds__(256) void k_pass_d(float* __restrict__ d2p,
                                                float* __restrict__ d2t) {
  __shared__ float tile[Dc * 32];
  int bb = blockIdx.x;
  float* arr = d2p;
  if (bb >= Bc * Hc * (Wc / 32)) { arr = d2t; bb -= Bc * Hc * (Wc / 32); }
  int b   = bb >> 9;              // H*(W/32) = 512 blocks per b
  int rem = bb & 511;
  int h   = rem >> 2;
  int w0  = (rem & 3) << 5;
  int tx  = threadIdx.x & 31;
  int ty  = threadIdx.x >> 5;
  int base = b * DHWc + h * Wc + w0;

  for (int r = ty; r < Dc; r += 8)
    lds_stage_f32(&arr[base + r * HWc + tx], &tile[r * 32 + tx]);
  lds_stage_wait();
  __syncthreads();

  for (int r = ty; r < Dc; r += 8) {
    float m = 3.0e38f;
#pragma unroll 4
    for (int j = 0; j < Dc; ++j) {
      float df = (float)(r - j);
      m = fminf(m, tile[j * 32 + tx] + df * df);
    }
    arr[base + r * HWc + tx] = m;
  }
}

// ---------------- Kernel 3: EDT pass along H (n=128, stride W) -------------
// Tile: 128 (h) x 32 (w). Blocks per array = B*D*(W/32) = 512.
__global__ __launch_bounds__(256) void k_pass_h(float* __restrict__ d2p,
                                                float* __restrict__ d2t) {
  __shared__ float tile[Hc * 32];
  int bb = blockIdx.x;
  float* arr = d2p;
  if (bb >= Bc * Dc * (Wc / 32)) { arr = d2t; bb -= Bc * Dc * (Wc / 32); }
  int bd = bb >> 2;               // combined b*D + d, 0..127
  int w0 = (bb & 3) << 5;
  int tx = threadIdx.x & 31;
  int ty = threadIdx.x >> 5;
  int base = bd * HWc + w0;

  for (int r = ty; r < Hc; r += 8)
    lds_stage_f32(&arr[base + r * Wc + tx], &tile[r * 32 + tx]);
  lds_stage_wait();
  __syncthreads();

  for (int r = ty; r < Hc; r += 8) {
    float m = 3.0e38f;
#pragma unroll 4
    for (int j = 0; j < Hc; ++j) {
      float df = (float)(r - j);
      m = fminf(m, tile[j * 32 + tx] + df * df);
    }
    arr[base + r * Wc + tx] = m;
  }
}

// ---------------- Kernel 4: EDT pass along W (n=128, stride 1) -------------
// One 128-wide line per block. Blocks per array = B*D*H = 16384.
__global__ __launch_bounds__(128) void k_pass_w(float* __restrict__ d2p,
                                                float* __restrict__ d2t) {
  __shared__ float line[Wc];
  int bb = blockIdx.x;
  float* arr = d2p;
  if (bb >= Bc * Dc * Hc) { arr = d2t; bb -= Bc * Dc * Hc; }
  int base = bb * Wc;
  int t = threadIdx.x;

  lds_stage_f32(&arr[base + t], &line[t]);
  lds_stage_wait();
  __syncthreads();

  float m = 3.0e38f;
#pragma unroll 4
  for (int j = 0; j < Wc; ++j) {
    float df = (float)(t - j);
    m = fminf(m, line[j] + df * df);
  }
  arr[base + t] = m;
}

// ---------------- Kernel 5: smooth-L1 partial sums -------------------------
__global__ __launch_bounds__(256) void k_loss(const float* __restrict__ d2p,
                                              const float* __restrict__ d2t,
                                              float* __restrict__ partial) {
  __shared__ float red[256];
  int t = threadIdx.x;
  int base = blockIdx.x * 2048;   // NTOT / 1024 blocks
  float s = 0.0f;
#pragma unroll
  for (int i = 0; i < 8; ++i) {
    int idx = base + i * 256 + t;
    float sp = sqrtf(d2p[idx]);
    float st = sqrtf(d2t[idx]);
    float d  = sp - st;
    float ad = fabsf(d);
    s += (ad < 1.0f) ? 0.5f * d * d : (ad - 0.5f);
  }
  red[t] = s;
  __syncthreads();
  for (int off = 128; off > 0; off >>= 1) {
    if (t < off) red[t] += red[t + off];
    __syncthreads();
  }
  if (t == 0) partial[blockIdx.x] = red[0];
}

// ---------------- Kernel 6: final reduce via v_wmma_f32_16x16x4_f32 --------
// B = ones  =>  D[m,n] = sum_k A[m,k] + C[m,n].  Summing all 256 C entries
// over the 16 replicated columns gives 16 * grand_total; scale by 2^-25 exact.
typedef float v2f __attribute__((ext_vector_type(2)));
typedef float v8f __attribute__((ext_vector_type(8)));

#if defined(__gfx1250__) && __has_builtin(__builtin_amdgcn_wmma_f32_16x16x4_f32)
#define HAS_WMMA_F32 1
#else
#define HAS_WMMA_F32 0
#endif

__global__ __launch_bounds__(32) void k_final(const float* __restrict__ partial,
                                              float* __restrict__ out) {
  int lane = threadIdx.x;
#if HAS_WMMA_F32
  v8f c = {};
  v2f bm = {1.0f, 1.0f};
  for (int ch = 0; ch < 1024; ch += 64) {
    int idx = ch + lane * 2;
    v2f a = {partial[idx], partial[idx + 1]};
    c = __builtin_amdgcn_wmma_f32_16x16x4_f32(
        /*neg_a=*/false, a, /*neg_b=*/false, bm,
        /*c_mod=*/(short)0, c, /*reuse_a=*/false, /*reuse_b=*/false);
  }
  float s = c[0] + c[1] + c[2] + c[3] + c[4] + c[5] + c[6] + c[7];
  for (int off = 16; off > 0; off >>= 1) s += __shfl_xor(s, off, 32);
  // s == 16 * total;  mean = total / NTOT = s / (16*NTOT) = s * 2^-25 (exact scale)
  if (lane == 0) out[0] = s * (1.0f / 33554432.0f);
#else
  float s = 0.0f;
  for (int i = lane; i < 1024; i += 32) s += partial[i];
  for (int off = 16; off > 0; off >>= 1) s += __shfl_xor(s, off, 32);
  if (lane == 0) out[0] = s * (1.0f / (float)NTOT);
#endif
}

// ---------------- Host launcher --------------------------------------------
extern "C" void kernel_launch(void* const* d_in, const int* in_sizes, int n_in,
                              void* d_out, int out_size, void* d_ws, size_t ws_size,
                              hipStream_t stream) {
  const float* pred   = (const float*)d_in[0];
  const float* target = (const float*)d_in[1];

  float* d2p     = (float*)d_ws;           // NTOT floats (8 MB)
  float* d2t     = d2p + NTOT;             // NTOT floats (8 MB)
  float* partial = d2t + NTOT;             // 1024 floats

  k_init  <<<NTOT / 256, 256, 0, stream>>>(pred, target, d2p, d2t);
  k_pass_d<<<2 * Bc * Hc * (Wc / 32), 256, 0, stream>>>(d2p, d2t);  // 2048
  k_pass_h<<<2 * Bc * Dc * (Wc / 32), 256, 0, stream>>>(d2p, d2t);  // 1024
  k_pass_w<<<2 * Bc * Dc * Hc,        128, 0, stream>>>(d2p, d2t);  // 32768
  k_loss  <<<1024, 256, 0, stream>>>(d2p, d2t, partial);
  k_final <<<1, 32, 0, stream>>>(partial, (float*)d_out);
}